// TropicalAttention_23888608101009
// MI455X (gfx1250) — compile-verified
//
#include <hip/hip_runtime.h>
#include <math.h>

#define DM   512
#define NH   8
#define DKK  64
#define B_   2
#define LL   256
#define ROWS (B_*LL)   /* 512 rows of x / context */

typedef __attribute__((ext_vector_type(2))) float v2f;
typedef __attribute__((ext_vector_type(8))) float v8f;

// ---------------------------------------------------------------------------
// C[512x512] = A[512x512] @ W^T, optional epilogue log1p(relu(.)) - lam[col].
// One wave32 computes one 16x16 tile via V_WMMA_F32_16X16X4_F32, K-step 4.
// A frag (16x4 f32): lanes 0-15 m=0..15 k={0,1}; lanes 16-31 m=0..15 k={2,3}.
// B frag (4x16 f32): symmetric, n = lane&15, k = 2*(lane>>4)+vgpr.
// C/D (8 VGPRs): vgpr i -> row 16*tr + 8*half + i, col = 16*tc + (lane&15).
// ---------------------------------------------------------------------------
__global__ __launch_bounds__(256)
void gemm512_wmma(const float* __restrict__ A, const float* __restrict__ W,
                  const float* __restrict__ lam, float* __restrict__ Y, int act)
{
    const int wave = threadIdx.x >> 5;
    const int lane = threadIdx.x & 31;
    const int tile = blockIdx.x * 8 + wave;      // 1024 tiles = 32x32
    const int tr   = tile >> 5;
    const int tc   = tile & 31;
    const int half = lane >> 4;
    const int idx  = lane & 15;

    const float* arow = A + (size_t)(tr * 16 + idx) * DM + 2 * half;
    const float* wrow = W + (size_t)(tc * 16 + idx) * DM + 2 * half;

    v8f acc = {0.f, 0.f, 0.f, 0.f, 0.f, 0.f, 0.f, 0.f};
    #pragma unroll 4
    for (int k = 0; k < DM; k += 4) {
        v2f a = *(const v2f*)(arow + k);
        v2f b = *(const v2f*)(wrow + k);
        acc = __builtin_amdgcn_wmma_f32_16x16x4_f32(
                  /*neg_a=*/false, a, /*neg_b=*/false, b,
                  /*c_mod=*/(short)0, acc, /*reuse_a=*/false, /*reuse_b=*/false);
    }

    const int mbase = tr * 16 + half * 8;
    const int col   = tc * 16 + idx;
    const float lm  = act ? lam[col] : 0.f;
    #pragma unroll
    for (int i = 0; i < 8; ++i) {
        float v = acc[i];
        if (act) v = log1pf(fmaxf(v, 0.f)) - lm;
        Y[(size_t)(mbase + i) * DM + col] = v;
    }
}

// ---------------------------------------------------------------------------
// Tropical linear per head: out[bh,l,o] = max_i( proj[b,l,h*64+i] + Wt[o,i] ).
// Block: 256 threads = 4 rows x 64 outputs. Wt staged in LDS, stride 65 so
// lane-varying 'o' hits distinct banks.
// ---------------------------------------------------------------------------
__global__ __launch_bounds__(256)
void tropical_linear(const float* __restrict__ proj, const float* __restrict__ Wt,
                     float* __restrict__ out)
{
    __shared__ float wl[DKK * 65];
    __shared__ float xl[4 * DKK];

    const int t  = threadIdx.x;
    const int bh = blockIdx.x >> 6;          // 16 head-batches
    const int lt = blockIdx.x & 63;          // 64 tiles of 4 rows
    const int b  = bh >> 3, h = bh & 7;

    #pragma unroll
    for (int r = 0; r < 16; ++r) {           // 4096 W elems / 256 thr
        int e = t + 256 * r;
        wl[(e >> 6) * 65 + (e & 63)] = Wt[e];
    }
    const int lr = t >> 6, oi = t & 63;
    xl[t] = proj[(size_t)(b * LL + lt * 4 + lr) * DM + h * DKK + oi];
    __syncthreads();

    const float* xr = xl + lr * DKK;         // broadcast reads within wave
    const float* wr = wl + oi * 65;          // conflict-free (stride 65)
    float m = -__builtin_inff();
    #pragma unroll 8
    for (int i = 0; i < DKK; ++i)
        m = fmaxf(m, xr[i] + wr[i]);
    out[(size_t)bh * LL * DKK + (lt * 4 + lr) * DKK + oi] = m;
}

// ---------------------------------------------------------------------------
// Tropical metric attention core. Block = (head bh, 16-row i-tile).
// K/V chunks of 64 rows in LDS (stride 65). scores[i,j] = min_d - max_d of
// (q[i,d]-k[j,d]); context[i,d] = max_j(scores[i,j] + v[j,d]).
// Writes scores to d_out region 2, expm1(context) head-merged to workspace.
// ---------------------------------------------------------------------------
__global__ __launch_bounds__(256)
void trop_attn(const float* __restrict__ qt, const float* __restrict__ kt,
               const float* __restrict__ vt, float* __restrict__ scores,
               float* __restrict__ ctxout)
{
    __shared__ float qs[16 * DKK];
    __shared__ float ks[64 * 65];
    __shared__ float vs[64 * 65];
    __shared__ float ss[16 * 64];

    const int t  = threadIdx.x;
    const int it = blockIdx.x;               // i-tile 0..15
    const int bh = blockIdx.y;               // 0..15

    const float* qp = qt + ((size_t)bh * LL + it * 16) * DKK;
    #pragma unroll
    for (int r = 0; r < 4; ++r) qs[t + 256 * r] = qp[t + 256 * r];

    const float NI = -__builtin_inff();
    float ctx[4] = {NI, NI, NI, NI};

    for (int jc = 0; jc < 4; ++jc) {
        const float* kp = kt + ((size_t)bh * LL + jc * 64) * DKK;
        const float* vp = vt + ((size_t)bh * LL + jc * 64) * DKK;
        __syncthreads();                     // previous chunk fully consumed
        #pragma unroll
        for (int r = 0; r < 16; ++r) {
            int e = t + 256 * r; int j = e >> 6, d = e & 63;
            ks[j * 65 + d] = kp[e];
            vs[j * 65 + d] = vp[e];
        }
        __syncthreads();

        // scores: 16 i x 64 j cells, 4 per thread
        #pragma unroll
        for (int r = 0; r < 4; ++r) {
            int c = t + 256 * r; int i = c >> 6, jj = c & 63;
            const float* qr = qs + i * DKK;      // broadcast within wave
            const float* kr = ks + jj * 65;      // conflict-free
            float mx = NI, mn = __builtin_inff();
            #pragma unroll 8
            for (int d = 0; d < DKK; ++d) {
                float df = qr[d] - kr[d];
                mx = fmaxf(mx, df);
                mn = fminf(mn, df);
            }
            float s = mn - mx;                   // == -(max - min)
            ss[c] = s;
            scores[(size_t)bh * LL * LL + (size_t)(it * 16 + i) * LL + jc * 64 + jj] = s;
        }
        __syncthreads();

        // running max-plus context: cell c -> (i = c/64, d = c%64)
        #pragma unroll
        for (int r = 0; r < 4; ++r) {
            int c = t + 256 * r; int i = c >> 6, d = c & 63;
            const float* sr = ss + i * 64;       // broadcast within wave
            float m = ctx[r];
            #pragma unroll 8
            for (int jj = 0; jj < 64; ++jj)
                m = fmaxf(m, sr[jj] + vs[jj * 65 + d]);
            ctx[r] = m;
        }
    }

    const int b = bh >> 3, h = bh & 7;
    #pragma unroll
    for (int r = 0; r < 4; ++r) {
        int c = t + 256 * r; int i = c >> 6, d = c & 63;
        ctxout[(size_t)(b * LL + it * 16 + i) * DM + h * DKK + d] = expm1f(ctx[r]);
    }
}

// ---------------------------------------------------------------------------
extern "C" void kernel_launch(void* const* d_in, const int* in_sizes, int n_in,
                              void* d_out, int out_size, void* d_ws, size_t ws_size,
                              hipStream_t stream)
{
    (void)in_sizes; (void)n_in; (void)out_size; (void)ws_size;
    const float* x   = (const float*)d_in[0];
    const float* Wq  = (const float*)d_in[1];
    const float* Wk  = (const float*)d_in[2];
    const float* Wv  = (const float*)d_in[3];
    const float* Wo  = (const float*)d_in[4];
    const float* lam = (const float*)d_in[5];
    const float* Wqt = (const float*)d_in[6];
    const float* Wkt = (const float*)d_in[7];
    const float* Wvt = (const float*)d_in[8];

    float* out = (float*)d_out;                 // (2,256,512) = 262144 f32
    float* sc  = out + (size_t)ROWS * DM;       // (16,256,256) = 1048576 f32

    const size_t N = (size_t)ROWS * DM;         // 262144 floats per region
    float* ws  = (float*)d_ws;
    float* qp  = ws;                            // Q projection
    float* kp  = ws + N;                        // K projection
    float* vp  = ws + 2 * N;                    // V projection
    float* qtb = ws + 3 * N;                    // tropical Q (16,256,64)
    float* ktb = ws + 4 * N;
    float* vtb = ws + 5 * N;
    float* ctx = qp;                            // reuse dead Q-proj region

    dim3 blk(256);
    gemm512_wmma<<<128, blk, 0, stream>>>(x, Wq, lam, qp, 1);
    gemm512_wmma<<<128, blk, 0, stream>>>(x, Wk, lam, kp, 1);
    gemm512_wmma<<<128, blk, 0, stream>>>(x, Wv, lam, vp, 1);

    tropical_linear<<<1024, blk, 0, stream>>>(qp, Wqt, qtb);
    tropical_linear<<<1024, blk, 0, stream>>>(kp, Wkt, ktb);
    tropical_linear<<<1024, blk, 0, stream>>>(vp, Wvt, vtb);

    trop_attn<<<dim3(16, 16), blk, 0, stream>>>(qtb, ktb, vtb, sc, ctx);

    gemm512_wmma<<<128, blk, 0, stream>>>(ctx, Wo, (const float*)nullptr, out, 0);
}